// BatchSpectralLoss_49924699849080
// MI455X (gfx1250) — compile-verified
//
#include <hip/hip_runtime.h>
#include <hip/hip_bf16.h>

typedef __attribute__((ext_vector_type(2))) float v2f;
typedef __attribute__((ext_vector_type(8))) float v8f;

#define LS_EPS 0.1f
#define ROWSPLIT 8

// ---------------------------------------------------------------------------
// Kernel 1: column partial sums via V_WMMA_F32_16X16X4_F32 with A == ones.
// D = ones(16x4) x B(4x16) + C  =>  every row of D is the column-sum of B.
// Each wave owns one 16-column chunk over rowsPerSplit rows.
// B layout (32-bit, 4x16): lanes 0-15 hold {K=0,K=1} at N=lane,
//                          lanes 16-31 hold {K=2,K=3} at N=lane-16.
// (With A all-ones, only the N->lane mapping matters; K-permutation is inert.)
// ---------------------------------------------------------------------------
__global__ void bsl_colsum_wmma(const float* __restrict__ A,
                                float* __restrict__ colpart,
                                int C, int rowsPerSplit, int nColChunks) {
  const int wave = blockIdx.x * (blockDim.x >> 5) + (threadIdx.x >> 5);
  const int lane = threadIdx.x & 31;
  const int chunk = wave % nColChunks;
  const int rs    = wave / nColChunks;
  const int colL  = chunk * 16 + (lane & 15);
  const int rowOff = (lane >> 4) * 2;            // 0 for lanes 0-15, 2 for 16-31

  v2f a; a.x = 1.0f; a.y = 1.0f;                 // all-ones A matrix
  v8f acc = {};

  const float* base = A + (size_t)(rs * rowsPerSplit) * C + colL;
  for (int k = 0; k < rowsPerSplit; k += 4) {
    const float* p = base + (size_t)(k + rowOff) * C;
    __builtin_prefetch(p + (size_t)4 * C, 0, 1); // next K-step tile
    v2f b;
    b.x = p[0];
    b.y = p[C];
    acc = __builtin_amdgcn_wmma_f32_16x16x4_f32(
        /*neg_a=*/false, a, /*neg_b=*/false, b,
        /*c_mod=*/(short)0, acc, /*reuse_a=*/false, /*reuse_b=*/false);
  }
  // D row M=0 lives in acc VGPR0, lanes 0-15 (N = lane).
  if (lane < 16) colpart[(size_t)rs * C + colL] = acc[0];
}

// ---------------------------------------------------------------------------
// Kernel 2: per-row label-smoothed CE pieces + row sum-of-squares.
// One 256-thread block per row; row (32 KB) stays L2/L1-resident for pass 2.
// ---------------------------------------------------------------------------
__global__ void bsl_row_ce(const float* __restrict__ A,
                           const int* __restrict__ pids,
                           float* __restrict__ rowloss,
                           float* __restrict__ rowsq,
                           int C) {
  __shared__ float r0[256];
  __shared__ float r1[256];
  __shared__ float r2[256];
  const int row = blockIdx.x;
  const int tid = threadIdx.x;
  const float* x = A + (size_t)row * C;

  // pass 1: row max
  float m = -__builtin_inff();
  for (int c = tid; c < C; c += 256) m = fmaxf(m, x[c]);
  r0[tid] = m; __syncthreads();
  for (int s = 128; s > 0; s >>= 1) {
    if (tid < s) r0[tid] = fmaxf(r0[tid], r0[tid + s]);
    __syncthreads();
  }
  m = r0[0];
  __syncthreads();

  // pass 2: Z = sum exp(x-m), srow = sum x, sq = sum x^2
  float z = 0.0f, srow = 0.0f, sq = 0.0f;
  for (int c = tid; c < C; c += 256) {
    float v = x[c];
    z += __expf(v - m);
    srow += v;
    sq = fmaf(v, v, sq);
  }
  r0[tid] = z; r1[tid] = srow; r2[tid] = sq;
  __syncthreads();
  for (int s = 128; s > 0; s >>= 1) {
    if (tid < s) {
      r0[tid] += r0[tid + s];
      r1[tid] += r1[tid + s];
      r2[tid] += r2[tid + s];
    }
    __syncthreads();
  }

  if (tid == 0) {
    float logZ = __logf(r0[0]);
    float srt  = r1[0];
    float xt   = x[pids[row]];
    float li = (1.0f - LS_EPS) * (xt - m - logZ)
             + (LS_EPS / (float)C) * srt
             - LS_EPS * (m + logZ);
    rowloss[row] = -li;
    rowsq[row]   = r2[0];
  }
}

// ---------------------------------------------------------------------------
// Kernel 3: finalize. dot(s,s) from ROWSPLIT partials per column, reduce row
// losses & row sumsqs, combine into the scalar output.
// ---------------------------------------------------------------------------
__global__ void bsl_finalize(const float* __restrict__ colpart,
                             const float* __restrict__ rowloss,
                             const float* __restrict__ rowsq,
                             float* __restrict__ out,
                             int N, int C) {
  __shared__ float r0[256];
  __shared__ float r1[256];
  __shared__ float r2[256];
  const int tid = threadIdx.x;

  float dss = 0.0f;
  for (int c = tid; c < C; c += 256) {
    float s = 0.0f;
    for (int k = 0; k < ROWSPLIT; ++k) s += colpart[(size_t)k * C + c];
    dss = fmaf(s, s, dss);
  }
  float lsum = 0.0f, qsum = 0.0f;
  for (int r = tid; r < N; r += 256) {
    lsum += rowloss[r];
    qsum += rowsq[r];
  }
  r0[tid] = dss; r1[tid] = lsum; r2[tid] = qsum;
  __syncthreads();
  for (int s = 128; s > 0; s >>= 1) {
    if (tid < s) {
      r0[tid] += r0[tid + s];
      r1[tid] += r1[tid + s];
      r2[tid] += r2[tid + s];
    }
    __syncthreads();
  }
  if (tid == 0) {
    // penalty = dot(s,s) - sum(A*A);  xloss = mean(rowloss)
    out[0] = (r0[0] - r2[0]) + r1[0] / (float)N;
  }
}

extern "C" void kernel_launch(void* const* d_in, const int* in_sizes, int n_in,
                              void* d_out, int out_size, void* d_ws, size_t ws_size,
                              hipStream_t stream) {
  const float* logits = (const float*)d_in[0];
  const int*   pids   = (const int*)d_in[1];
  float* out = (float*)d_out;
  float* ws  = (float*)d_ws;

  const int N = in_sizes[1];              // 4096
  const int C = in_sizes[0] / N;          // 8192

  const int rowsPerSplit = N / ROWSPLIT;  // 512 rows per wave (128 WMMA steps)
  const int nColChunks   = C / 16;        // 512 column chunks

  float* colpart = ws;                                  // ROWSPLIT * C
  float* rowloss = ws + (size_t)ROWSPLIT * C;           // N
  float* rowsq   = rowloss + N;                         // N

  const int totalWaves = nColChunks * ROWSPLIT;         // 4096 waves
  dim3 g1(totalWaves / 8);                              // 8 waves / block
  bsl_colsum_wmma<<<g1, 256, 0, stream>>>(logits, colpart, C, rowsPerSplit, nColChunks);
  bsl_row_ce<<<N, 256, 0, stream>>>(logits, pids, rowloss, rowsq, C);
  bsl_finalize<<<1, 256, 0, stream>>>(colpart, rowloss, rowsq, out, N, C);
}